// PVNet_58944131170391
// MI455X (gfx1250) — compile-verified
//
#include <hip/hip_runtime.h>

#define NUM_POINTS 16
#define PATCH      5
#define BB         4
#define NV         6
#define CTOT       428

typedef __attribute__((ext_vector_type(2))) float v2f;
typedef __attribute__((ext_vector_type(8))) float v8f;

// ---------------------------------------------------------------------------
// Kernel 1: project the 16 line samples through all (n,b) projection matrices
// with V_WMMA_F32_16X16X4_F32.
//   A (16x4): row m = homogeneous point m = points3d[b] + (m/15)*vecs3d[b]
//   B (4x16): col j<3 = row j of proj[n][b] (proj transposed), cols 3..15 = 0
//   D (16x16): D[m][j] = projected coordinate j (x,y,z) of point m
// Lane layouts per CDNA5 ISA 7.12.2 (f32, wave32):
//   A: VGPR v, lane L: M = L&15, K = v + 2*(L>>4)
//   B: VGPR v, lane L: N = L&15, K = v + 2*(L>>4)
//   D: VGPR r, lane L: M = r + 8*(L>>4), N = L&15
// ---------------------------------------------------------------------------
__global__ __launch_bounds__(32) void pv_project_wmma(
    const float* __restrict__ proj,      // (N,B,3,4)
    const float* __restrict__ points3d,  // (B,4,1)
    const float* __restrict__ vecs3d,    // (B,4,1)
    float* __restrict__ lineXY)          // (16,B,N,2) float pixel coords
{
  __shared__ float lds[NUM_POINTS * 3];
  const int lane = threadIdx.x;      // 0..31, one full wave (EXEC all ones)
  const int half = lane >> 4;        // 0 or 1
  const int lo   = lane & 15;
  const int k0   = 0 + 2 * half;
  const int k1   = 1 + 2 * half;

  for (int b = 0; b < BB; ++b) {
    // A-matrix: 16 points x 4 homogeneous coords
    v2f a;
    {
      const float step = (float)lo * (1.0f / (float)(NUM_POINTS - 1));
      a.x = points3d[b * 4 + k0] + step * vecs3d[b * 4 + k0];
      a.y = points3d[b * 4 + k1] + step * vecs3d[b * 4 + k1];
    }
    for (int n = 0; n < NV; ++n) {
      // B-matrix: proj[n][b] transposed into cols 0..2, zero elsewhere
      const float* pnb = proj + ((size_t)n * BB + b) * 12;
      v2f bm;
      bm.x = (lo < 3) ? pnb[lo * 4 + k0] : 0.0f;
      bm.y = (lo < 3) ? pnb[lo * 4 + k1] : 0.0f;

      v8f c = {};
      // (neg_a, A, neg_b, B, c_mod, C, reuse_a, reuse_b)
      v8f d = __builtin_amdgcn_wmma_f32_16x16x4_f32(
          false, a, false, bm, (short)0, c, false, false);

      // Scatter the 3 useful columns of D into LDS as coords[m][j]
      if (lo < 3) {
#pragma unroll
        for (int r = 0; r < 8; ++r) {
          lds[(r + 8 * half) * 3 + lo] = d[r];
        }
      }
      __syncthreads();
      if (lane < NUM_POINTS) {
        const float x = lds[lane * 3 + 0];
        const float y = lds[lane * 3 + 1];
        const float z = lds[lane * 3 + 2];
        const float inv = 1.0f / z;
        float* o = lineXY + ((size_t)(lane * BB + b) * NV + n) * 2;
        o[0] = x * inv;
        o[1] = y * inv;
      }
      __syncthreads();
    }
  }
}

// ---------------------------------------------------------------------------
// Kernel 2: gather 5x5 patches at the truncated per-scale indices for all 6
// views and compute the unbiased variance across views.
// One thread per output element; flat id = (((b*428+c)*16+p)*5+py)*5+px,
// which is exactly the reference's (B,428,num,P,P) output order.
// Loads are made UNCONDITIONAL (coords clamped into range, result zeroed by
// cndmask when out of range) so all 6 gathers issue back-to-back and one
// s_wait_loadcnt covers them — this kernel is gather-latency bound.
// ---------------------------------------------------------------------------
__global__ __launch_bounds__(256) void pv_patchvar(
    const float* __restrict__ f0, const float* __restrict__ f1,
    const float* __restrict__ f2, const float* __restrict__ f3,
    const float* __restrict__ f4,
    const float* __restrict__ lineXY,   // (16,B,N,2)
    float* __restrict__ out, int total)
{
  const int id = blockIdx.x * blockDim.x + threadIdx.x;
  if (id >= total) return;

  const int px = id % PATCH;
  const int py = (id / PATCH) % PATCH;
  const int p  = (id / (PATCH * PATCH)) % NUM_POINTS;
  const int c  = (id / (PATCH * PATCH * NUM_POINTS)) % CTOT;
  const int b  =  id / (PATCH * PATCH * NUM_POINTS * CTOT);

  // channel -> pyramid level (chans 16,38,61,128,185; strides 2,4,8,16,32)
  const float* feat; int C, H, W, cl; float inv_s;
  if (c < 16)       { feat = f0; C = 16;  H = 256; W = 320; inv_s = 1.f/2.f;  cl = c;       }
  else if (c < 54)  { feat = f1; C = 38;  H = 128; W = 160; inv_s = 1.f/4.f;  cl = c - 16;  }
  else if (c < 115) { feat = f2; C = 61;  H = 64;  W = 80;  inv_s = 1.f/8.f;  cl = c - 54;  }
  else if (c < 243) { feat = f3; C = 128; H = 32;  W = 40;  inv_s = 1.f/16.f; cl = c - 115; }
  else              { feat = f4; C = 185; H = 16;  W = 20;  inv_s = 1.f/32.f; cl = c - 243; }

  const v2f* lxy = (const v2f*)(lineXY + (size_t)(p * BB + b) * NV * 2);
  const int dy = py - (PATCH / 2);
  const int dx = px - (PATCH / 2);

  // Unconditional gathers: clamp address into range, mask value afterwards.
  float vals[NV];
  bool  ok[NV];
#pragma unroll
  for (int n = 0; n < NV; ++n) {
    const v2f xy = lxy[n];                    // one b64 load per view
    // truncation toward zero matches .astype(int32) on positive coords;
    // strides are powers of two so *inv_s is bit-exact vs /s
    const int xi  = (int)(xy.x * inv_s);
    const int yi  = (int)(xy.y * inv_s);
    const int row = yi + dy;
    const int col = xi + dx;
    ok[n] = (row >= 0) & (row < H) & (col >= 0) & (col < W);
    const int rc = min(max(row, 0), H - 1);
    const int cc = min(max(col, 0), W - 1);
    vals[n] = feat[(((size_t)(b * NV + n) * C + cl) * H + rc) * W + cc];
  }

  float sum = 0.0f, sumsq = 0.0f;
#pragma unroll
  for (int n = 0; n < NV; ++n) {
    const float v = ok[n] ? vals[n] : 0.0f;   // v_cndmask, no branch
    sum   += v;
    sumsq  = fmaf(v, v, sumsq);
  }
  const float mean = sum * (1.0f / (float)NV);
  out[id] = (sumsq - sum * mean) * (1.0f / (float)(NV - 1));
}

extern "C" void kernel_launch(void* const* d_in, const int* in_sizes, int n_in,
                              void* d_out, int out_size, void* d_ws, size_t ws_size,
                              hipStream_t stream) {
  const float* f0   = (const float*)d_in[0];
  const float* f1   = (const float*)d_in[1];
  const float* f2   = (const float*)d_in[2];
  const float* f3   = (const float*)d_in[3];
  const float* f4   = (const float*)d_in[4];
  const float* proj = (const float*)d_in[5];
  const float* pts  = (const float*)d_in[6];
  const float* vecs = (const float*)d_in[7];

  float* lineXY = (float*)d_ws;   // 16*4*6*2 = 768 floats (3 KB) of scratch

  pv_project_wmma<<<1, 32, 0, stream>>>(proj, pts, vecs, lineXY);

  const int total = BB * CTOT * NUM_POINTS * PATCH * PATCH;  // 684,800
  pv_patchvar<<<(total + 255) / 256, 256, 0, stream>>>(
      f0, f1, f2, f3, f4, lineXY, (float*)d_out, total);
}